// PseudoNMS_73598559584661
// MI455X (gfx1250) — compile-verified
//
#include <hip/hip_runtime.h>
#include <stdint.h>

// PseudoNMS: out = heatmap * (maxpool3==h) * (maxpool5==h) * (maxpool7==h)
// Nested windows => product of masks == (maxpool7 == h).  -inf padding for a
// max-pool is equivalent to clamp-to-edge sampling.  Memory-bound: 378 MB of
// mandatory traffic -> ~16 us at 23.3 TB/s on MI455X.  Use CDNA5 async
// global->LDS copies (ASYNCcnt) to stage tiles, separable max in LDS.

#define IMG_H   192
#define IMG_W   192
#define TILE    64
#define HALO    3
#define IN_T    (TILE + 2 * HALO)   // 70
#define THREADS 256

// ---------- CDNA5 async global->LDS copy (gfx1250) -------------------------
typedef __attribute__((address_space(1))) int GlobInt;
typedef __attribute__((address_space(3))) int LdsInt;

__device__ __forceinline__ void async_copy_b32(const float* plane,
                                               uint32_t byte_off,
                                               float* lds_ptr) {
#if __has_builtin(__builtin_amdgcn_global_load_async_to_lds_b32)
  __builtin_amdgcn_global_load_async_to_lds_b32(
      (GlobInt*)(uintptr_t)((const char*)plane + byte_off),
      (LdsInt*)(uint32_t)(uintptr_t)lds_ptr,
      /*offset=*/0, /*cpol=*/0);
#else
  const uint32_t lds_off = (uint32_t)(uintptr_t)lds_ptr;
  asm volatile("global_load_async_to_lds_b32 %0, %1, %2 offset:0"
               :
               : "v"(lds_off), "v"(byte_off), "s"(plane)
               : "memory");
#endif
}

__device__ __forceinline__ void wait_async_all() {
#if __has_builtin(__builtin_amdgcn_s_wait_asynccnt)
  __builtin_amdgcn_s_wait_asynccnt(0);
#else
  asm volatile("s_wait_asynccnt 0x0" ::: "memory");
#endif
}

__device__ __forceinline__ float max7(const float* w) {
  return fmaxf(fmaxf(fmaxf(w[0], w[1]), fmaxf(w[2], w[3])),
               fmaxf(fmaxf(w[4], w[5]), w[6]));
}

__global__ __launch_bounds__(THREADS) void pseudo_nms7_kernel(
    const float* __restrict__ src, float* __restrict__ dst) {
  __shared__ float s_in[IN_T * IN_T];   // 70x70 clamped input tile (19.6 KB)
  __shared__ float s_h[IN_T * TILE];    // 70x64 horizontal 7-max   (17.9 KB)

  const int tid = threadIdx.x;
  const int x0 = blockIdx.x * TILE;
  const int y0 = blockIdx.y * TILE;
  const float* plane_src = src + (size_t)blockIdx.z * (IMG_H * IMG_W);
  float* plane_dst       = dst + (size_t)blockIdx.z * (IMG_H * IMG_W);

  // ---- Phase A: async-stage clamped 70x70 tile into LDS -------------------
  for (int idx = tid; idx < IN_T * IN_T; idx += THREADS) {
    const int ty = idx / IN_T;
    const int tx = idx - ty * IN_T;
    int gy = y0 - HALO + ty;
    int gx = x0 - HALO + tx;
    gy = gy < 0 ? 0 : (gy > IMG_H - 1 ? IMG_H - 1 : gy);
    gx = gx < 0 ? 0 : (gx > IMG_W - 1 ? IMG_W - 1 : gx);
    async_copy_b32(plane_src, (uint32_t)((gy * IMG_W + gx) * 4), &s_in[idx]);
  }
  wait_async_all();
  __syncthreads();

  // ---- Phase B: horizontal 7-wide sliding max (register-tiled, 16 px) -----
  // 70 rows x 4 chunks = 280 tasks.  Bank-conflict-free: row stride 70.
  for (int task = tid; task < IN_T * (TILE / 16); task += THREADS) {
    const int row = task >> 2;
    const int cx = (task & 3) * 16;
    const float* r = &s_in[row * IN_T + cx];
    float w[16 + 2 * HALO];
#pragma unroll
    for (int i = 0; i < 16 + 2 * HALO; ++i) w[i] = r[i];
    float* hout = &s_h[row * TILE + cx];
#pragma unroll
    for (int i = 0; i < 16; ++i) hout[i] = max7(&w[i]);
  }
  __syncthreads();

  // ---- Phase C: vertical 7-tall sliding max + compare + store -------------
  // 64 cols x 4 row-chunks = 256 tasks -> exactly one per thread.
  {
    const int col = tid & (TILE - 1);
    const int ry  = (tid >> 6) * 16;
    float w[16 + 2 * HALO];
#pragma unroll
    for (int i = 0; i < 16 + 2 * HALO; ++i) w[i] = s_h[(ry + i) * TILE + col];
#pragma unroll
    for (int i = 0; i < 16; ++i) {
      const float m = max7(&w[i]);
      const float c = s_in[(ry + i + HALO) * IN_T + (col + HALO)];
      const float o = (m == c) ? c : 0.0f;
      plane_dst[(size_t)(y0 + ry + i) * IMG_W + (x0 + col)] = o;
    }
  }
}

extern "C" void kernel_launch(void* const* d_in, const int* in_sizes, int n_in,
                              void* d_out, int out_size, void* d_ws, size_t ws_size,
                              hipStream_t stream) {
  const float* src = (const float*)d_in[0];
  float* dst = (float*)d_out;
  const int planes = in_sizes[0] / (IMG_H * IMG_W);  // 16*80 = 1280
  dim3 grid(IMG_W / TILE, IMG_H / TILE, planes);     // 3 x 3 x 1280
  pseudo_nms7_kernel<<<grid, dim3(THREADS), 0, stream>>>(src, dst);
}